// VectorQuantizerEMA_39900246180015
// MI455X (gfx1250) — compile-verified
//
#include <hip/hip_runtime.h>
#include <hip/hip_bf16.h>
#include <math.h>
#include <stdint.h>

typedef __attribute__((ext_vector_type(2))) float v2f;
typedef __attribute__((ext_vector_type(8))) float v8f;
typedef __attribute__((ext_vector_type(4))) unsigned int v4u;
typedef __attribute__((ext_vector_type(4))) int v4i;
typedef __attribute__((ext_vector_type(8))) int v8i;

#define NUM_K 1024
#define DIM 256
#define NROWS 32768           // 32*32*32
#define HWSZ 1024             // 32*32
#define DECAY_C 0.99f
#define ONE_M_DECAY 0.01f
#define EPS_C 1e-5f
#define COMMIT 0.25f
#define TOTAL_ELEMS 8388608   // 32*256*32*32

// workspace layout (float units)
#define WS_COUNTS   32768     // float[1024]     (idx int[32768] occupies [0,32768))
#define WS_DW       33792     // float[262144]
#define WS_SSE      295936    // float[1]
#define WS_NTOT     295937    // float[1]
#define WS_CNORM    295938    // float[1024]

#define BROW 260              // padded LDS row stride (floats): 256 + 4 (TDM pad_amount=3 -> 4 dwords)
#define TILE_BYTES (16 * BROW * 4)   // 16640 B per 16x256 tile

__global__ void k_zero(float* p, int n) {
  int i = blockIdx.x * blockDim.x + threadIdx.x;
  if (i < n) p[i] = 0.0f;
}

__global__ void k_cnorm(const float* __restrict__ cb, float* __restrict__ cnorm) {
  __shared__ float red[256];
  int k = blockIdx.x;
  float v = cb[k * DIM + threadIdx.x];
  red[threadIdx.x] = v * v;
  __syncthreads();
  for (int s = 128; s > 0; s >>= 1) {
    if (threadIdx.x < s) red[threadIdx.x] += red[threadIdx.x + s];
    __syncthreads();
  }
  if (threadIdx.x == 0) cnorm[k] = red[0];
}

// TDM: load a 16-row x 256-float tile (row stride 256 floats in memory) into LDS
// with 4-dword padding after every 256 dwords -> LDS row stride = 260 floats.
__device__ __forceinline__ void tdm_load_tile(const float* gsrc, unsigned lds_off) {
  unsigned long long ga = (unsigned long long)(uintptr_t)gsrc;
  v4u g0;
  g0.x = 1u;                                            // count=1, user mode, no gather
  g0.y = lds_off;                                       // lds_addr
  g0.z = (unsigned)(ga & 0xffffffffu);                  // global_addr[31:0]
  g0.w = (unsigned)((ga >> 32) & 0x01ffffffu) | (2u << 30);  // global_addr[56:32] | type=2
  v8i g1;
  g1[0] = (int)((2u << 16)      // data_size = 4 bytes
              | (1u << 20)      // pad_enable
              | (7u << 22)      // pad_interval: 256 dwords
              | (3u << 25));    // pad_amount: 4 dwords
  g1[1] = (int)(256u << 16);    // tensor_dim0 = 256 (bits 63:48)
  g1[2] = (int)(16u << 16);     // tensor_dim1 = 16  (bits 95:80)
  g1[3] = (int)(256u << 16);    // tile_dim0 = 256   (bits 127:112)
  g1[4] = 16;                   // tile_dim1 = 16    (bits 143:128)
  g1[5] = 256;                  // tensor_dim0_stride = 256 (bits 191:160)
  g1[6] = 0;
  g1[7] = 0;
  v4i gz = {0, 0, 0, 0};
#if defined(__clang_major__) && (__clang_major__ >= 23)
  v8i gz8 = {0, 0, 0, 0, 0, 0, 0, 0};
  __builtin_amdgcn_tensor_load_to_lds(g0, g1, gz, gz, gz8, 0);
#else
  __builtin_amdgcn_tensor_load_to_lds(g0, g1, gz, gz, 0);
#endif
}

// 2 waves/block; each wave owns 16 rows (n). A fragments hoisted to registers,
// B tiles DMA'd by the Tensor Data Mover into double-buffered LDS.
__global__ __launch_bounds__(64) void k_argmin(const float* __restrict__ z,
                                               const float* __restrict__ cb,
                                               const float* __restrict__ cnorm,
                                               int* __restrict__ idx,
                                               float* __restrict__ counts) {
  // phase 1: per-wave A staging (16x260 each); phase 2: B double buffer (16x260 each)
  __shared__ float lds[2 * 16 * BROW];   // 33,280 B

  const int tid  = threadIdx.x;
  const int wv   = tid >> 5;
  const int lane = tid & 31;
  const int lo   = lane & 15;
  const int hi   = lane >> 4;

  float* Aw = lds + wv * (16 * BROW);
  {
    // stage A: z_flat[n][d] = z[b*262144 + d*1024 + hw]
    const int n0w = blockIdx.x * 32 + wv * 16;
    const int b   = n0w >> 10;
    const int hw0 = n0w & 1023;
    const float* zb = z + (size_t)b * (DIM * HWSZ) + hw0;
    #pragma unroll 4
    for (int j = 0; j < 128; ++j) {
      int d = j * 2 + hi;
      Aw[lo * BROW + d] = zb[(size_t)d * HWSZ + lo];
    }
  }
  __syncthreads();

  // hoist all A fragments into registers (invariant over k-tiles)
  v2f areg[64];
  {
    const float* Abase = Aw + lo * BROW + hi * 2;
    #pragma unroll
    for (int s = 0; s < 64; ++s) areg[s] = *(const v2f*)(Abase + 4 * s);
  }
  // drain DS before the TDM engine may overwrite this LDS region
  asm volatile("s_wait_dscnt 0" ::: "memory");
  __syncthreads();

  const unsigned ldsbase = (unsigned)(size_t)(void*)lds;
  if (wv == 0) tdm_load_tile(cb, ldsbase);   // tile 0 -> buffer 0

  float best[8];
  int   bidx[8];
  #pragma unroll
  for (int j = 0; j < 8; ++j) { best[j] = 3.4e38f; bidx[j] = 0; }

  for (int kt = 0; kt < NUM_K / 16; ++kt) {
    if (wv == 0) __builtin_amdgcn_s_wait_tensorcnt(0);   // tile kt arrived
    __syncthreads();                                      // visible to all waves; prev buf free
    if (wv == 0 && kt + 1 < NUM_K / 16)                   // prefetch tile kt+1
      tdm_load_tile(cb + (size_t)(kt + 1) * 16 * DIM,
                    ldsbase + (unsigned)(((kt + 1) & 1) * TILE_BYTES));

    const float* Bbase = lds + (kt & 1) * (16 * BROW) + lo * BROW + hi * 2;

    v8f acc0 = {0.f, 0.f, 0.f, 0.f, 0.f, 0.f, 0.f, 0.f};
    v8f acc1 = {0.f, 0.f, 0.f, 0.f, 0.f, 0.f, 0.f, 0.f};
    #pragma unroll
    for (int s = 0; s < 32; ++s) {                        // two independent chains
      v2f b0 = *(const v2f*)(Bbase + 8 * s);
      v2f b1 = *(const v2f*)(Bbase + 8 * s + 4);
      acc0 = __builtin_amdgcn_wmma_f32_16x16x4_f32(false, areg[2 * s],     false, b0,
                                                   (short)0, acc0, false, false);
      acc1 = __builtin_amdgcn_wmma_f32_16x16x4_f32(false, areg[2 * s + 1], false, b1,
                                                   (short)0, acc1, false, false);
    }

    const int k0 = kt * 16;
    float cn = cnorm[k0 + lo];
    int   kk = k0 + lo;
    #pragma unroll
    for (int j = 0; j < 8; ++j) {
      float sc = cn - 2.0f * (acc0[j] + acc1[j]);  // ||c||^2 - 2 z.c
      if (sc < best[j]) { best[j] = sc; bidx[j] = kk; }
    }
  }

  // argmin across the 16 lanes of each half-wave (tie-break: lowest index)
  #pragma unroll
  for (int m = 1; m < 16; m <<= 1) {
    #pragma unroll
    for (int j = 0; j < 8; ++j) {
      float os = __shfl_xor(best[j], m, 32);
      int   oi = __shfl_xor(bidx[j], m, 32);
      if (os < best[j] || (os == best[j] && oi < bidx[j])) { best[j] = os; bidx[j] = oi; }
    }
  }

  if (lo == 0) {
    const int nb = blockIdx.x * 32 + wv * 16 + hi * 8;  // C layout: M = j + 8*hi
    #pragma unroll
    for (int j = 0; j < 8; ++j) {
      idx[nb + j] = bidx[j];
      atomicAdd(&counts[bidx[j]], 1.0f);
    }
  }
}

// One block per (d, n-chunk): LDS histogram for dw column, coalesced z/qout traffic.
__global__ __launch_bounds__(256) void k_quant(const float* __restrict__ z,
                                               const float* __restrict__ cb,
                                               const int* __restrict__ idx,
                                               float* __restrict__ qout,
                                               float* __restrict__ dw,
                                               float* __restrict__ sse) {
  __shared__ float ldw[NUM_K];
  __shared__ float red[256];
  const int d = blockIdx.x & 255;        // dim slice
  const int g = blockIdx.x >> 8;         // n-chunk 0..3
  const int t = threadIdx.x;
  for (int i = t; i < NUM_K; i += 256) ldw[i] = 0.0f;
  __syncthreads();

  float lsse = 0.0f;
  const float* cbd = cb + d;
  for (int it = 0; it < 32; ++it) {
    int n  = g * 8192 + it * 256 + t;    // 0..32767
    int b  = n >> 10;
    int hw = n & 1023;
    int k  = idx[n];
    size_t off = (size_t)b * (DIM * HWSZ) + (size_t)d * HWSZ + hw;
    float zv = z[off];
    float q  = cbd[(size_t)k * DIM];
    qout[off] = q;                       // straight-through value == quantized
    atomicAdd(&ldw[k], zv);              // LDS float atomic (ds_add_f32)
    float df = zv - q;
    lsse += df * df;
  }
  __syncthreads();

  red[t] = lsse;
  __syncthreads();
  for (int s = 128; s > 0; s >>= 1) {
    if (t < s) red[t] += red[t + s];
    __syncthreads();
  }
  if (t == 0) atomicAdd(sse, red[0]);

  for (int k = t; k < NUM_K; k += 256)
    if (ldw[k] != 0.0f) atomicAdd(&dw[(size_t)k * DIM + d], ldw[k]);
}

__global__ void k_fin1(const float* __restrict__ counts,
                       const float* __restrict__ ema_cs,
                       const float* __restrict__ sse,
                       float* __restrict__ ncs_out,
                       float* __restrict__ loss_out,
                       float* __restrict__ perp_out,
                       float* __restrict__ ntot) {
  __shared__ float r1[1024];
  __shared__ float r2[1024];
  int k = threadIdx.x;
  float cnt = counts[k];
  float ncs = ema_cs[k] * DECAY_C + ONE_M_DECAY * cnt;
  ncs_out[k] = ncs;
  float p = cnt / (float)NROWS;
  r1[k] = ncs;
  r2[k] = p * logf(p + 1e-10f);
  __syncthreads();
  for (int s = 512; s > 0; s >>= 1) {
    if (k < s) { r1[k] += r1[k + s]; r2[k] += r2[k + s]; }
    __syncthreads();
  }
  if (k == 0) {
    ntot[0]     = r1[0];
    loss_out[0] = COMMIT * (sse[0] / (float)TOTAL_ELEMS);
    perp_out[0] = expf(-r2[0]);
  }
}

__global__ void k_fin2(const float* __restrict__ ncs_out,
                       const float* __restrict__ ema_dw,
                       const float* __restrict__ dw,
                       const float* __restrict__ ntot,
                       float* __restrict__ cb_out,
                       float* __restrict__ edw_out) {
  int k = blockIdx.x;
  int d = threadIdx.x;
  float n   = ntot[0];
  float ncs = ncs_out[k];
  float cs  = (ncs + EPS_C) / (n + (float)NUM_K * EPS_C) * n;
  float ndw = ema_dw[k * DIM + d] * DECAY_C + ONE_M_DECAY * dw[k * DIM + d];
  edw_out[k * DIM + d] = ndw;
  cb_out[k * DIM + d]  = ndw / cs;
}

extern "C" void kernel_launch(void* const* d_in, const int* in_sizes, int n_in,
                              void* d_out, int out_size, void* d_ws, size_t ws_size,
                              hipStream_t stream) {
  const float* z      = (const float*)d_in[0];
  const float* cb     = (const float*)d_in[1];
  const float* ema_cs = (const float*)d_in[2];
  const float* ema_dw = (const float*)d_in[3];

  float* out      = (float*)d_out;
  float* q_out    = out;                    // 8,388,608
  float* loss_out = out + 8388608;          // 1
  float* perp_out = out + 8388609;          // 1
  float* cb_out   = out + 8388610;          // 262,144
  float* ncs_out  = out + 8650754;          // 1,024
  float* edw_out  = out + 8651778;          // 262,144

  float* wf     = (float*)d_ws;
  int*   idx    = (int*)d_ws;
  float* counts = wf + WS_COUNTS;
  float* dw     = wf + WS_DW;
  float* sse    = wf + WS_SSE;
  float* ntot   = wf + WS_NTOT;
  float* cnorm  = wf + WS_CNORM;

  // counts + dw + sse are contiguous: zero them every call
  int zcount = 1024 + 262144 + 1;
  k_zero<<<(zcount + 255) / 256, 256, 0, stream>>>(counts, zcount);
  k_cnorm<<<NUM_K, 256, 0, stream>>>(cb, cnorm);
  k_argmin<<<NROWS / 32, 64, 0, stream>>>(z, cb, cnorm, idx, counts);
  k_quant<<<4 * DIM, 256, 0, stream>>>(z, cb, idx, q_out, dw, sse);
  k_fin1<<<1, 1024, 0, stream>>>(counts, ema_cs, sse, ncs_out, loss_out, perp_out, ntot);
  k_fin2<<<NUM_K, 256, 0, stream>>>(ncs_out, ema_dw, dw, ntot, cb_out, edw_out);
}